// GIN_12352325943894
// MI455X (gfx1250) — compile-verified
//
#include <hip/hip_runtime.h>

#define NN 50000
#define NE 400000
#define DD 300
#define NL 5
#define CC 128
#define NGG 256
#define EPS 1e-5f

typedef float v2f __attribute__((ext_vector_type(2)));
typedef float v8f __attribute__((ext_vector_type(8)));

// ---------------------------------------------------------------- copy h -> z
__global__ void gin_copy4(const float4* __restrict__ src, float4* __restrict__ dst, int n4) {
    int i = blockIdx.x * blockDim.x + threadIdx.x;
    if (i < n4) dst[i] = src[i];
}

// ----------------------------------------------- z[dst] += h[src] over edges
__global__ void gin_edge_agg(const float* __restrict__ h,
                             const long long* __restrict__ esrc,
                             const long long* __restrict__ edst,
                             float* __restrict__ z) {
    int e = blockIdx.x;
    int s = (int)esrc[e];
    int d = (int)edst[e];
    const float* hs = h + (long)s * DD;
    float*       zd = z + (long)d * DD;
    for (int f = threadIdx.x; f < DD; f += blockDim.x)
        unsafeAtomicAdd(&zd[f], hs[f]);
}

// ------------------------------------------------- Out = A[NN,DD] @ W[DD,DD] + bias
// 256 threads = 8 waves (4 row-groups x 2 col-groups); block tile 128 rows x 64 cols.
// Each wave: 32x32 C tile = 4 x v8f accumulators; per K micro-step 4 ds_load_b64
// feed 4 independent v_wmma_f32_16x16x4_f32. K staged in 32-chunks via LDS with
// register-prefetch software pipeline.
__global__ __launch_bounds__(256)
void gin_gemm_wmma(const float* __restrict__ A,
                   const float* __restrict__ W,
                   const float* __restrict__ bias,
                   float* __restrict__ Out) {
    __shared__ float lds_a[128 * 36];   // [row][k], padded stride 36
    __shared__ float lds_b[64 * 34];    // [n][k] (transposed), padded stride 34
    const int row0 = blockIdx.x * 128;
    const int n0   = blockIdx.y * 64;
    const int tid  = threadIdx.x;
    const int wave = tid >> 5;
    const int lane = tid & 31;
    const int rowg = wave & 3;          // 4 row-groups of 32 rows
    const int colg = wave >> 2;         // 2 col-groups of 32 cols
    const int m     = lane & 15;
    const int khalf = (lane >> 4) << 1;

    // A-staging coords: thread -> (row, 16-wide k half), 4 float4 each
    const int ar  = tid >> 1;
    const int akk = (tid & 1) * 16;
    // B-staging coords: thread -> (k row, 8 cols), 2 float4 each
    const int bk  = tid >> 3;
    const int bc  = (tid & 7) * 8;

    v8f acc00 = {0.f,0.f,0.f,0.f,0.f,0.f,0.f,0.f};
    v8f acc01 = acc00, acc10 = acc00, acc11 = acc00;

    float4 areg[4];
    float4 breg[2];

    auto gload = [&](int k0) {
        const int grow = row0 + ar;
        #pragma unroll
        for (int j = 0; j < 4; ++j) {
            int kg = k0 + akk + 4 * j;
            float4 v = make_float4(0.f, 0.f, 0.f, 0.f);
            if (grow < NN && kg < DD)
                v = *reinterpret_cast<const float4*>(A + (long)grow * DD + kg);
            areg[j] = v;
        }
        const int kg = k0 + bk;
        #pragma unroll
        for (int j = 0; j < 2; ++j) {
            int cg = n0 + bc + 4 * j;
            float4 v = make_float4(0.f, 0.f, 0.f, 0.f);
            if (kg < DD && cg < DD)
                v = *reinterpret_cast<const float4*>(W + (long)kg * DD + cg);
            breg[j] = v;
        }
    };

    auto lstore = [&]() {
        #pragma unroll
        for (int j = 0; j < 4; ++j)
            *reinterpret_cast<float4*>(&lds_a[ar * 36 + akk + 4 * j]) = areg[j];
        const float* bs = reinterpret_cast<const float*>(breg);
        #pragma unroll
        for (int j = 0; j < 8; ++j)
            lds_b[(bc + j) * 34 + bk] = bs[j];
    };

    gload(0);
    const int NCHUNK = (DD + 31) / 32;   // 10
    for (int chunk = 0; chunk < NCHUNK; ++chunk) {
        lstore();
        __syncthreads();
        if (chunk + 1 < NCHUNK) gload((chunk + 1) * 32);   // prefetch overlaps compute

        const float* pa0 = &lds_a[(rowg * 32 + m) * 36];
        const float* pa1 = pa0 + 16 * 36;
        const float* pb0 = &lds_b[(colg * 32 + m) * 34];
        const float* pb1 = pb0 + 16 * 34;
        #pragma unroll
        for (int s = 0; s < 8; ++s) {
            int kk = 4 * s + khalf;
            v2f a0 = *reinterpret_cast<const v2f*>(pa0 + kk);
            v2f a1 = *reinterpret_cast<const v2f*>(pa1 + kk);
            v2f b0 = *reinterpret_cast<const v2f*>(pb0 + kk);
            v2f b1 = *reinterpret_cast<const v2f*>(pb1 + kk);
            acc00 = __builtin_amdgcn_wmma_f32_16x16x4_f32(false, a0, false, b0, (short)0, acc00, false, false);
            acc01 = __builtin_amdgcn_wmma_f32_16x16x4_f32(false, a0, false, b1, (short)0, acc01, false, false);
            acc10 = __builtin_amdgcn_wmma_f32_16x16x4_f32(false, a1, false, b0, (short)0, acc10, false, false);
            acc11 = __builtin_amdgcn_wmma_f32_16x16x4_f32(false, a1, false, b1, (short)0, acc11, false, false);
        }
        __syncthreads();
    }

    // epilogue: C layout — VGPR r: M = r + 8*(lane>>4), N = lane&15
    const int mhi   = (lane >> 4) << 3;
    const int rbase = row0 + rowg * 32;
    const int gcol0 = n0 + colg * 32 + m;
    const int gcol1 = gcol0 + 16;
    if (gcol0 < DD) {
        float bv = bias[gcol0];
        #pragma unroll
        for (int r = 0; r < 8; ++r) {
            int g0 = rbase + mhi + r;
            int g1 = g0 + 16;
            if (g0 < NN) Out[(long)g0 * DD + gcol0] = acc00[r] + bv;
            if (g1 < NN) Out[(long)g1 * DD + gcol0] = acc10[r] + bv;
        }
    }
    if (gcol1 < DD) {
        float bv = bias[gcol1];
        #pragma unroll
        for (int r = 0; r < 8; ++r) {
            int g0 = rbase + mhi + r;
            int g1 = g0 + 16;
            if (g0 < NN) Out[(long)g0 * DD + gcol1] = acc01[r] + bv;
            if (g1 < NN) Out[(long)g1 * DD + gcol1] = acc11[r] + bv;
        }
    }
}

// ------------------------------------------------- column sum / sumsq (LDS-staged)
#define SROWS 200
__global__ void gin_bn_stats(const float* __restrict__ t,
                             float* __restrict__ gsum, float* __restrict__ gsq) {
    __shared__ float ls[DD];
    __shared__ float lq[DD];
    for (int i = threadIdx.x; i < DD; i += blockDim.x) { ls[i] = 0.f; lq[i] = 0.f; }
    __syncthreads();
    long base = (long)blockIdx.x * SROWS * DD;   // multiple of DD
    for (int e = threadIdx.x; e < SROWS * DD; e += blockDim.x) {
        long idx = base + e;
        if (idx < (long)NN * DD) {
            float v = t[idx];
            int f = e % DD;
            unsafeAtomicAdd(&ls[f], v);
            unsafeAtomicAdd(&lq[f], v * v);
        }
    }
    __syncthreads();
    for (int i = threadIdx.x; i < DD; i += blockDim.x) {
        unsafeAtomicAdd(&gsum[i], ls[i]);
        unsafeAtomicAdd(&gsq[i],  lq[i]);
    }
}

__global__ void gin_bn_finalize(const float* __restrict__ gsum, const float* __restrict__ gsq,
                                float* __restrict__ mean, float* __restrict__ rstd) {
    int i = threadIdx.x + blockIdx.x * blockDim.x;
    if (i < DD) {
        float mu  = gsum[i] * (1.0f / NN);
        float var = gsq[i] * (1.0f / NN) - mu * mu;   // biased, matches jnp var
        mean[i] = mu;
        rstd[i] = rsqrtf(var + EPS);
    }
}

__global__ void gin_bn_apply_relu(const float* __restrict__ in, float* __restrict__ out,
                                  const float* __restrict__ mean, const float* __restrict__ rstd,
                                  const float* __restrict__ g, const float* __restrict__ b) {
    long idx = (long)blockIdx.x * blockDim.x + threadIdx.x;
    if (idx < (long)NN * DD) {
        int f = (int)(idx % DD);
        float v = (in[idx] - mean[f]) * rstd[f] * g[f] + b[f];
        out[idx] = fmaxf(v, 0.0f);
    }
}

// ------------------------------------------------- per-graph pooling (sums)
__global__ void gin_pool(const float* __restrict__ h, const long long* __restrict__ batch,
                         float* __restrict__ pooled) {
    long idx = (long)blockIdx.x * blockDim.x + threadIdx.x;
    if (idx < (long)NN * DD) {
        int n = (int)(idx / DD);
        int f = (int)(idx - (long)n * DD);
        int gidx = (int)batch[n];
        unsafeAtomicAdd(&pooled[(long)gidx * DD + f], h[idx]);
    }
}

__global__ void gin_counts(const long long* __restrict__ batch, float* __restrict__ counts) {
    int n = blockIdx.x * blockDim.x + threadIdx.x;
    if (n < NN) unsafeAtomicAdd(&counts[(int)batch[n]], 1.0f);
}

// ------------------------------------------------- readout
__global__ void gin_readout(const float* __restrict__ pooled,   // [6][NGG][DD] (sums)
                            const float* __restrict__ counts,
                            const float* __restrict__ fcW,      // [6][DD][CC]
                            const float* __restrict__ fcb,      // [6][CC]
                            float* __restrict__ out) {
    int gi = blockIdx.x;
    int c  = threadIdx.x;
    float inv = 1.0f / fmaxf(counts[gi], 1.0f);
    float acc = 0.0f;
    for (int i = 0; i < NL + 1; ++i) {
        const float* p = pooled + ((long)i * NGG + gi) * DD;
        const float* w = fcW + (long)i * DD * CC + c;
        float dot = 0.0f;
        for (int d = 0; d < DD; ++d) dot += p[d] * w[(long)d * CC];
        acc += dot * inv + fcb[i * CC + c];
    }
    out[(long)gi * CC + c] = acc;
}

// ------------------------------------------------------------------ driver
extern "C" void kernel_launch(void* const* d_in, const int* in_sizes, int n_in,
                              void* d_out, int out_size, void* d_ws, size_t ws_size,
                              hipStream_t stream) {
    const float*     x      = (const float*)d_in[0];
    const long long* eidx   = (const long long*)d_in[1];
    const long long* batch  = (const long long*)d_in[2];
    const float*     mlpW1  = (const float*)d_in[3];
    const float*     mlpb1  = (const float*)d_in[4];
    const float*     mlpg1  = (const float*)d_in[5];
    const float*     mlpbe1 = (const float*)d_in[6];
    const float*     mlpW2  = (const float*)d_in[7];
    const float*     mlpb2  = (const float*)d_in[8];
    const float*     bng    = (const float*)d_in[9];
    const float*     bnb    = (const float*)d_in[10];
    const float*     fcW    = (const float*)d_in[11];
    const float*     fcb    = (const float*)d_in[12];
    float*           out    = (float*)d_out;

    const long ND = (long)NN * DD;
    float* h_buf  = (float*)d_ws;
    float* zbuf   = h_buf + ND;
    float* tbuf   = zbuf + ND;
    float* pooled = tbuf + ND;                           // (NL+1)*NGG*DD
    float* counts = pooled + (long)(NL + 1) * NGG * DD;  // NGG
    float* ssum   = counts + NGG;                        // DD
    float* ssq    = ssum + DD;                           // DD
    float* smean  = ssq + DD;                            // DD
    float* srstd  = smean + DD;                          // DD

    const long long* esrc = eidx;
    const long long* edst = eidx + NE;

    const int n4 = (int)(ND / 4);
    const int copy_blocks  = (n4 + 255) / 256;
    const int elem_blocks  = (int)((ND + 255) / 256);
    const int stats_blocks = (NN + SROWS - 1) / SROWS;
    dim3 gemm_grid((NN + 127) / 128, (DD + 63) / 64);    // (391, 5)

    // zero pooled sums + counts (contiguous)
    hipMemsetAsync(pooled, 0, ((size_t)(NL + 1) * NGG * DD + NGG) * sizeof(float), stream);

    gin_counts<<<(NN + 255) / 256, 256, 0, stream>>>(batch, counts);
    gin_pool<<<elem_blocks, 256, 0, stream>>>(x, batch, pooled);  // depth 0 = x

    for (int i = 0; i < NL; ++i) {
        const float* h = (i == 0) ? x : h_buf;
        const float* W1  = mlpW1  + (long)i * DD * DD;
        const float* b1  = mlpb1  + i * DD;
        const float* g1  = mlpg1  + i * DD;
        const float* be1 = mlpbe1 + i * DD;
        const float* W2  = mlpW2  + (long)i * DD * DD;
        const float* b2  = mlpb2  + i * DD;
        const float* og  = bng + i * DD;
        const float* ob  = bnb + i * DD;

        // z = h + scatter_add(h[src] -> dst)
        gin_copy4<<<copy_blocks, 256, 0, stream>>>((const float4*)h, (float4*)zbuf, n4);
        gin_edge_agg<<<NE, 128, 0, stream>>>(h, esrc, edst, zbuf);

        // t = z @ W1 + b1
        gin_gemm_wmma<<<gemm_grid, 256, 0, stream>>>(zbuf, W1, b1, tbuf);

        // BN(g1,be1) + ReLU, in place on t
        hipMemsetAsync(ssum, 0, 2 * DD * sizeof(float), stream);
        gin_bn_stats<<<stats_blocks, 256, 0, stream>>>(tbuf, ssum, ssq);
        gin_bn_finalize<<<1, 512, 0, stream>>>(ssum, ssq, smean, srstd);
        gin_bn_apply_relu<<<elem_blocks, 256, 0, stream>>>(tbuf, tbuf, smean, srstd, g1, be1);

        // u = t @ W2 + b2  (reuse z buffer)
        gin_gemm_wmma<<<gemm_grid, 256, 0, stream>>>(tbuf, W2, b2, zbuf);

        // outer BN + ReLU -> h_buf
        hipMemsetAsync(ssum, 0, 2 * DD * sizeof(float), stream);
        gin_bn_stats<<<stats_blocks, 256, 0, stream>>>(zbuf, ssum, ssq);
        gin_bn_finalize<<<1, 512, 0, stream>>>(ssum, ssq, smean, srstd);
        gin_bn_apply_relu<<<elem_blocks, 256, 0, stream>>>(zbuf, h_buf, smean, srstd, og, ob);

        gin_pool<<<elem_blocks, 256, 0, stream>>>(h_buf, batch, pooled + (long)(i + 1) * NGG * DD);
    }

    gin_readout<<<NGG, CC, 0, stream>>>(pooled, counts, fcW, fcb, out);
}